// AttentionPool2d_91225105367084
// MI455X (gfx1250) — compile-verified
//
#include <hip/hip_runtime.h>

typedef __attribute__((ext_vector_type(2))) float v2f;
typedef __attribute__((ext_vector_type(8))) float v8f;

#define B_  32
#define E_  512
#define S_  257
#define NH_ 8
#define HD_ 64

// ---------------------------------------------------------------------------
// fp32 WMMA 16x16x4: D = A(16x4) * B(4x16) + C  (wave32, all lanes active)
// ---------------------------------------------------------------------------
static __device__ __forceinline__ v8f wmma4(v2f a, v2f b, v8f c) {
  // (neg_a, A, neg_b, B, c_mod, C, reuse_a, reuse_b); NEG must be 0 for f32
  return __builtin_amdgcn_wmma_f32_16x16x4_f32(false, a, false, b, (short)0, c,
                                               false, false);
}

static __device__ __forceinline__ float wave_sum(float v) {
  #pragma unroll
  for (int off = 16; off > 0; off >>= 1) v += __shfl_xor(v, off, 32);
  return v;
}

// ---------------------------------------------------------------------------
// Kernel 1a: spatial mean per (b,e) over 256 pixels (real & imag)
// one wave per (b,e)
// ---------------------------------------------------------------------------
__global__ void mean_kernel(const float* __restrict__ xr,
                            const float* __restrict__ xi,
                            float* __restrict__ mr, float* __restrict__ mi) {
  int wave = blockIdx.x * 8 + (threadIdx.x >> 5);   // wave == b*E + e
  int lane = threadIdx.x & 31;
  const float* pr = xr + (size_t)wave * 256 + lane * 8;
  const float* pi = xi + (size_t)wave * 256 + lane * 8;
  float4 a = ((const float4*)pr)[0], b4 = ((const float4*)pr)[1];
  float4 c = ((const float4*)pi)[0], d4 = ((const float4*)pi)[1];
  float sr = a.x + a.y + a.z + a.w + b4.x + b4.y + b4.z + b4.w;
  float si = c.x + c.y + c.z + c.w + d4.x + d4.y + d4.z + d4.w;
  sr = wave_sum(sr);
  si = wave_sum(si);
  if (lane == 0) {
    mr[wave] = sr * (1.0f / 256.0f);
    mi[wave] = si * (1.0f / 256.0f);
  }
}

// ---------------------------------------------------------------------------
// Kernel 1b: xs[b,s,e] = (s==0 ? mean[b,e] : x[b,e,s-1]) + pos[e,s]
// row-major [B*S, E], coalesced stores
// ---------------------------------------------------------------------------
__global__ void build_xs(const float* __restrict__ xr, const float* __restrict__ xi,
                         const float* __restrict__ mr, const float* __restrict__ mi,
                         const float* __restrict__ pos_r, const float* __restrict__ pos_i,
                         float* __restrict__ xs_r, float* __restrict__ xs_i) {
  size_t t = (size_t)blockIdx.x * 256 + threadIdx.x;   // < B*S*E
  int e = (int)(t % E_);
  size_t rest = t / E_;
  int s = (int)(rest % S_);
  int b = (int)(rest / S_);
  float vr_, vi_;
  if (s == 0) {
    vr_ = mr[b * E_ + e];
    vi_ = mi[b * E_ + e];
  } else {
    size_t idx = ((size_t)(b * E_ + e)) * 256 + (s - 1);
    vr_ = xr[idx];
    vi_ = xi[idx];
  }
  vr_ += pos_r[e * S_ + s];
  vi_ += pos_i[e * S_ + s];
  xs_r[t] = vr_;
  xs_i[t] = vi_;
}

// ---------------------------------------------------------------------------
// Kernel 2: complex GEMM for K,V.
//   out[m, n] = sum_e xs[m, e] * w_in[512 + n, e]   (complex), n in [0,1024)
// One wave computes a 32(M) x 64(N) block: 2 m-tiles x 4 n-tiles,
// both real & imag accumulators -> 16 v8f accumulators, 32 WMMAs per k-step.
// real += xr*wr + (-xi)*wi ; imag += xr*wi + xi*wr
// ---------------------------------------------------------------------------
__global__ __launch_bounds__(256) void kv_gemm(
    const float* __restrict__ xs_r, const float* __restrict__ xs_i,
    const float* __restrict__ w_r,  const float* __restrict__ w_i,
    const float* __restrict__ bn_r, const float* __restrict__ bn_i,
    float* __restrict__ kr, float* __restrict__ ki,
    float* __restrict__ vr, float* __restrict__ vi) {
  int wave = blockIdx.x * 8 + (threadIdx.x >> 5);  // 257*16 = 4112 waves
  int mb = wave >> 4;                              // 0..256
  int nb = wave & 15;                              // 0..15
  int lane = threadIdx.x & 31;
  int half = lane >> 4;
  int l16 = lane & 15;
  size_t m0 = (size_t)mb * 32;
  int n0 = nb * 64;

  const float* a0r = xs_r + (m0 + l16) * E_ + half * 2;
  const float* a0i = xs_i + (m0 + l16) * E_ + half * 2;
  const float* a1r = a0r + (size_t)16 * E_;
  const float* a1i = a0i + (size_t)16 * E_;
  const float* brp[4];
  const float* bip[4];
  #pragma unroll
  for (int j = 0; j < 4; ++j) {
    int row = 512 + n0 + j * 16 + l16;     // K rows 512..1023, V rows 1024..1535
    brp[j] = w_r + (size_t)row * E_ + half * 2;
    bip[j] = w_i + (size_t)row * E_ + half * 2;
  }

  v8f cr0[4] = {}; v8f ci0[4] = {};
  v8f cr1[4] = {}; v8f ci1[4] = {};

  for (int k = 0; k < E_; k += 4) {
    v2f x0r = *(const v2f*)(a0r + k);
    v2f x0i = *(const v2f*)(a0i + k);
    v2f x1r = *(const v2f*)(a1r + k);
    v2f x1i = *(const v2f*)(a1i + k);
    v2f x0n = -x0i;
    v2f x1n = -x1i;
    #pragma unroll
    for (int j = 0; j < 4; ++j) {
      v2f wr_ = *(const v2f*)(brp[j] + k);
      v2f wi_ = *(const v2f*)(bip[j] + k);
      cr0[j] = wmma4(x0r, wr_, cr0[j]);
      cr0[j] = wmma4(x0n, wi_, cr0[j]);
      ci0[j] = wmma4(x0r, wi_, ci0[j]);
      ci0[j] = wmma4(x0i, wr_, ci0[j]);
      cr1[j] = wmma4(x1r, wr_, cr1[j]);
      cr1[j] = wmma4(x1n, wi_, cr1[j]);
      ci1[j] = wmma4(x1r, wi_, ci1[j]);
      ci1[j] = wmma4(x1i, wr_, ci1[j]);
    }
  }

  #pragma unroll
  for (int j = 0; j < 4; ++j) {
    int n = n0 + j * 16 + l16;
    float bR = bn_r[512 + n];
    float bI = bn_i[512 + n];
    float* outR; float* outI; int f;
    if (n < 512) { outR = kr; outI = ki; f = n; }
    else         { outR = vr; outI = vi; f = n - 512; }
    #pragma unroll
    for (int r = 0; r < 8; ++r) {
      size_t m = m0 + r + half * 8;        // C layout: vgpr r -> M = r (+8 hi half)
      outR[m * E_ + f] = cr0[j][r] + bR;
      outI[m * E_ + f] = ci0[j][r] + bI;
      size_t m2 = m + 16;
      outR[m2 * E_ + f] = cr1[j][r] + bR;
      outI[m2 * E_ + f] = ci1[j][r] + bI;
    }
  }
}

// ---------------------------------------------------------------------------
// Kernel 3a: q0[b, f] = xs[b, 0, :] . w_in[f, :]  (complex) + b_in[f], f<512
// one wave per (b, f); lanes split E into 16-element chunks
// ---------------------------------------------------------------------------
__global__ void q0_kernel(const float* __restrict__ xs_r, const float* __restrict__ xs_i,
                          const float* __restrict__ w_r,  const float* __restrict__ w_i,
                          const float* __restrict__ b_r,  const float* __restrict__ b_i,
                          float* __restrict__ q0r, float* __restrict__ q0i) {
  int wave = blockIdx.x * 8 + (threadIdx.x >> 5);  // 16384 waves
  int lane = threadIdx.x & 31;
  int b = wave >> 9;
  int f = wave & 511;
  const float* xr = xs_r + (size_t)b * S_ * E_ + lane * 16;
  const float* xi = xs_i + (size_t)b * S_ * E_ + lane * 16;
  const float* wr = w_r + (size_t)f * E_ + lane * 16;
  const float* wi = w_i + (size_t)f * E_ + lane * 16;
  float sr = 0.f, si = 0.f;
  #pragma unroll
  for (int j = 0; j < 4; ++j) {
    float4 a = ((const float4*)xr)[j];
    float4 c = ((const float4*)xi)[j];
    float4 u = ((const float4*)wr)[j];
    float4 v = ((const float4*)wi)[j];
    sr += a.x * u.x - c.x * v.x + a.y * u.y - c.y * v.y +
          a.z * u.z - c.z * v.z + a.w * u.w - c.w * v.w;
    si += a.x * v.x + c.x * u.x + a.y * v.y + c.y * u.y +
          a.z * v.z + c.z * u.z + a.w * v.w + c.w * u.w;
  }
  sr = wave_sum(sr);
  si = wave_sum(si);
  if (lane == 0) {
    q0r[b * E_ + f] = sr + b_r[f];
    q0i[b * E_ + f] = si + b_i[f];
  }
}

// ---------------------------------------------------------------------------
// Kernel 3b: per (b,h): logits over 257 keys, dual softmax (real/imag),
// pooled[b, h*64+d] = sum_s softmax_w[s] * V[b,s,h*64+d]   (complex)
// ---------------------------------------------------------------------------
__global__ void attn_kernel(const float* __restrict__ q0r, const float* __restrict__ q0i,
                            const float* __restrict__ kr,  const float* __restrict__ ki,
                            const float* __restrict__ vr,  const float* __restrict__ vi,
                            float* __restrict__ plr, float* __restrict__ pli) {
  __shared__ float qr_s[HD_], qi_s[HD_];
  __shared__ float lr_s[S_], li_s[S_];
  __shared__ float redA[256], redB[256];
  int t = threadIdx.x;
  int b = blockIdx.x >> 3;
  int h = blockIdx.x & 7;

  if (t < HD_) {                      // scale q by 1/sqrt(64)
    qr_s[t] = q0r[b * E_ + h * HD_ + t] * 0.125f;
    qi_s[t] = q0i[b * E_ + h * HD_ + t] * 0.125f;
  }
  __syncthreads();

  for (int s = t; s < S_; s += 256) {
    const float* kpr = kr + ((size_t)(b * S_ + s)) * E_ + h * HD_;
    const float* kpi = ki + ((size_t)(b * S_ + s)) * E_ + h * HD_;
    float sr = 0.f, si = 0.f;
    #pragma unroll
    for (int j = 0; j < 16; ++j) {
      float4 ar4 = ((const float4*)kpr)[j];
      float4 ai4 = ((const float4*)kpi)[j];
      int d = j * 4;
      sr += qr_s[d+0]*ar4.x - qi_s[d+0]*ai4.x;  si += qr_s[d+0]*ai4.x + qi_s[d+0]*ar4.x;
      sr += qr_s[d+1]*ar4.y - qi_s[d+1]*ai4.y;  si += qr_s[d+1]*ai4.y + qi_s[d+1]*ar4.y;
      sr += qr_s[d+2]*ar4.z - qi_s[d+2]*ai4.z;  si += qr_s[d+2]*ai4.z + qi_s[d+2]*ar4.z;
      sr += qr_s[d+3]*ar4.w - qi_s[d+3]*ai4.w;  si += qr_s[d+3]*ai4.w + qi_s[d+3]*ar4.w;
    }
    lr_s[s] = sr;
    li_s[s] = si;
  }
  __syncthreads();

  // --- max reduce (real & imag) ---
  float vR = lr_s[t];  if (t == 0) vR = fmaxf(vR, lr_s[256]);
  float vI = li_s[t];  if (t == 0) vI = fmaxf(vI, li_s[256]);
  redA[t] = vR; redB[t] = vI;
  __syncthreads();
  for (int o = 128; o > 0; o >>= 1) {
    if (t < o) { redA[t] = fmaxf(redA[t], redA[t + o]); redB[t] = fmaxf(redB[t], redB[t + o]); }
    __syncthreads();
  }
  float mR = redA[0], mI = redB[0];
  __syncthreads();

  // --- exp + sum reduce ---
  float sR = 0.f, sI = 0.f;
  for (int s = t; s < S_; s += 256) {
    float er = __expf(lr_s[s] - mR); lr_s[s] = er; sR += er;
    float ei = __expf(li_s[s] - mI); li_s[s] = ei; sI += ei;
  }
  redA[t] = sR; redB[t] = sI;
  __syncthreads();
  for (int o = 128; o > 0; o >>= 1) {
    if (t < o) { redA[t] += redA[t + o]; redB[t] += redB[t + o]; }
    __syncthreads();
  }
  float invR = 1.f / redA[0], invI = 1.f / redB[0];
  __syncthreads();
  for (int s = t; s < S_; s += 256) { lr_s[s] *= invR; li_s[s] *= invI; }
  __syncthreads();

  // --- pooled = sum_s w[s] * V[s]   (4 s-groups x 64 d-threads) ---
  int g = t >> 6, d = t & 63;
  float ar_ = 0.f, ai_ = 0.f;
  for (int s = g; s < S_; s += 4) {
    size_t idx = ((size_t)(b * S_ + s)) * E_ + h * HD_ + d;
    float vvr = vr[idx], vvi = vi[idx];
    float wwr = lr_s[s], wwi = li_s[s];
    ar_ += wwr * vvr - wwi * vvi;
    ai_ += wwr * vvi + wwi * vvr;
  }
  redA[t] = ar_; redB[t] = ai_;
  __syncthreads();
  if (t < 64) {
    float pr = redA[t] + redA[t + 64] + redA[t + 128] + redA[t + 192];
    float pi = redB[t] + redB[t + 64] + redB[t + 128] + redB[t + 192];
    plr[b * E_ + h * HD_ + t] = pr;
    pli[b * E_ + h * HD_ + t] = pi;
  }
}

// ---------------------------------------------------------------------------
// Kernel 4: small complex GEMM [32,512] x W[512,512]^T + bias, WMMA tiles.
// interleave=1 writes complex64 (r,i) pairs into outR (final output).
// ---------------------------------------------------------------------------
__global__ void proj_gemm(const float* __restrict__ Ar, const float* __restrict__ Ai,
                          const float* __restrict__ Wr, const float* __restrict__ Wi,
                          const float* __restrict__ br, const float* __restrict__ bi,
                          float* __restrict__ outR, float* __restrict__ outI,
                          int interleave) {
  int wave = blockIdx.x * 8 + (threadIdx.x >> 5);   // 64 waves
  int mt = wave >> 5;                               // 0..1
  int nt = wave & 31;                               // 0..31
  int lane = threadIdx.x & 31;
  int half = lane >> 4, l16 = lane & 15;
  const float* ar = Ar + (size_t)(mt * 16 + l16) * E_ + half * 2;
  const float* ai = Ai + (size_t)(mt * 16 + l16) * E_ + half * 2;
  const float* wr = Wr + (size_t)(nt * 16 + l16) * E_ + half * 2;
  const float* wi = Wi + (size_t)(nt * 16 + l16) * E_ + half * 2;
  v8f cr = {}; v8f ci = {};
  for (int k = 0; k < E_; k += 4) {
    v2f xr = *(const v2f*)(ar + k);
    v2f xi = *(const v2f*)(ai + k);
    v2f xn = -xi;
    v2f br_ = *(const v2f*)(wr + k);
    v2f bi_ = *(const v2f*)(wi + k);
    cr = wmma4(xr, br_, cr);
    cr = wmma4(xn, bi_, cr);
    ci = wmma4(xr, bi_, ci);
    ci = wmma4(xi, br_, ci);
  }
  int n = nt * 16 + l16;
  float bR = br[n], bI = bi[n];
  #pragma unroll
  for (int r = 0; r < 8; ++r) {
    int m = mt * 16 + r + half * 8;
    float oR = cr[r] + bR, oI = ci[r] + bI;
    if (interleave) {
      size_t o = 2 * ((size_t)m * E_ + n);
      outR[o] = oR;
      outR[o + 1] = oI;
    } else {
      outR[(size_t)m * E_ + n] = oR;
      outI[(size_t)m * E_ + n] = oI;
    }
  }
}

// ---------------------------------------------------------------------------
extern "C" void kernel_launch(void* const* d_in, const int* in_sizes, int n_in,
                              void* d_out, int out_size, void* d_ws, size_t ws_size,
                              hipStream_t stream) {
  (void)in_sizes; (void)n_in; (void)out_size; (void)ws_size;
  const float* x_real  = (const float*)d_in[0];
  const float* x_imag  = (const float*)d_in[1];
  const float* pos_r   = (const float*)d_in[2];
  const float* pos_i   = (const float*)d_in[3];
  const float* w_in_r  = (const float*)d_in[4];
  const float* w_in_i  = (const float*)d_in[5];
  const float* b_in_r  = (const float*)d_in[6];
  const float* b_in_i  = (const float*)d_in[7];
  const float* w_out_r = (const float*)d_in[8];
  const float* w_out_i = (const float*)d_in[9];
  const float* b_out_r = (const float*)d_in[10];
  const float* b_out_i = (const float*)d_in[11];
  const float* w_p_r   = (const float*)d_in[12];
  const float* w_p_i   = (const float*)d_in[13];
  const float* b_p_r   = (const float*)d_in[14];
  const float* b_p_i   = (const float*)d_in[15];

  float* ws = (float*)d_ws;
  const size_t nSE = (size_t)B_ * S_ * E_;   // 4,210,688
  const size_t nBE = (size_t)B_ * E_;        // 16,384
  float* xs_r = ws;            float* xs_i = xs_r + nSE;
  float* kr   = xs_i + nSE;    float* ki   = kr + nSE;
  float* vr   = ki + nSE;      float* vi   = vr + nSE;
  float* mr   = vi + nSE;      float* mi   = mr + nBE;
  float* q0r  = mi + nBE;      float* q0i  = q0r + nBE;
  float* plr  = q0i + nBE;     float* pli  = plr + nBE;
  float* a0r  = pli + nBE;     float* a0i  = a0r + nBE;

  mean_kernel<<<2048, 256, 0, stream>>>(x_real, x_imag, mr, mi);
  build_xs<<<16448, 256, 0, stream>>>(x_real, x_imag, mr, mi, pos_r, pos_i, xs_r, xs_i);
  kv_gemm<<<514, 256, 0, stream>>>(xs_r, xs_i, w_in_r, w_in_i, b_in_r, b_in_i,
                                   kr, ki, vr, vi);
  q0_kernel<<<2048, 256, 0, stream>>>(xs_r, xs_i, w_in_r, w_in_i, b_in_r, b_in_i,
                                      q0r, q0i);
  attn_kernel<<<256, 256, 0, stream>>>(q0r, q0i, kr, ki, vr, vi, plr, pli);
  proj_gemm<<<8, 256, 0, stream>>>(plr, pli, w_out_r, w_out_i, b_out_r, b_out_i,
                                   a0r, a0i, 0);
  proj_gemm<<<8, 256, 0, stream>>>(a0r, a0i, w_p_r, w_p_i, b_p_r, b_p_i,
                                   (float*)d_out, nullptr, 1);
}